// GraphAttentionLayer_38517266711068
// MI455X (gfx1250) — compile-verified
//
#include <hip/hip_runtime.h>
#include <hip/hip_bf16.h>
#include <stdint.h>

// GAT layer, MI455X (gfx1250, wave32).
// Roofline: adj (256 MB) read exactly once at 23.3 TB/s ~ 11 us; everything else
// (Wh 2MB, f1/f2 32KB) is L2-resident. The 8.6 GFLOP P@Wh product is done with
// V_WMMA_F32_16X16X4_F32 so it hides under the memory time. The 8192^2
// attention matrix is never materialized: mask + leaky_relu + softmax + PV are
// fused into one pass using a global-max shift (valid because leaky_relu is
// monotone, and softmax is shift-invariant).

#define N_NODES 8192
#define IN_F    512
#define OUT_F   64
#define ALPHA   0.2f

typedef __attribute__((ext_vector_type(2))) float v2f;
typedef __attribute__((ext_vector_type(8))) float v8f;

__device__ __forceinline__ v8f wmma_f32(v2f a, v2f b, v8f c) {
  // D(16x16) = A(16x4) * B(4x16) + C, fp32 matrix pipe
  return __builtin_amdgcn_wmma_f32_16x16x4_f32(
      /*neg_a=*/false, a, /*neg_b=*/false, b,
      /*c_mod=*/(short)0, c, /*reuse_a=*/false, /*reuse_b=*/false);
}

// ---------------------------------------------------------------------------
// Kernel 1: Wh = h @ W   (8192x512 @ 512x64), one wave per 16-row tile.
// A layout (16x4 f32): lanes 0-15 hold M=lane, VGPR v <-> K=v (+2 for hi half).
// B layout (4x16 f32): VGPR v <-> K=v (lanes 0-15) / K=v+2 (lanes 16-31).
// C layout (16x16 f32): VGPR v <-> M=v (+8 for hi half), N=lane&15.
// ---------------------------------------------------------------------------
__global__ void __launch_bounds__(256)
wh_gemm_kernel(const float* __restrict__ h, const float* __restrict__ W,
               float* __restrict__ Wh) {
  const int lane = threadIdx.x & 31;
  const int wave = threadIdx.x >> 5;
  const int r  = lane & 15;
  const int hh = lane >> 4;
  const int m0 = (blockIdx.x * 8 + wave) * 16;

  v8f acc0 = {}, acc1 = {}, acc2 = {}, acc3 = {};
  const float* hrow = h + (size_t)(m0 + r) * IN_F + 2 * hh;

#pragma unroll 4
  for (int k = 0; k < IN_F; k += 4) {
    v2f a = *(const v2f*)(hrow + k);
    const float* wk = W + (size_t)(k + 2 * hh) * OUT_F + r;
    v2f b0, b1, b2, b3;
    b0.x = wk[0];  b0.y = wk[OUT_F + 0];
    b1.x = wk[16]; b1.y = wk[OUT_F + 16];
    b2.x = wk[32]; b2.y = wk[OUT_F + 32];
    b3.x = wk[48]; b3.y = wk[OUT_F + 48];
    acc0 = wmma_f32(a, b0, acc0);
    acc1 = wmma_f32(a, b1, acc1);
    acc2 = wmma_f32(a, b2, acc2);
    acc3 = wmma_f32(a, b3, acc3);
  }

  float* orow = Wh + (size_t)m0 * OUT_F + r;
#pragma unroll
  for (int v = 0; v < 8; ++v) {
    const size_t row = (size_t)(v + 8 * hh) * OUT_F;
    orow[row + 0]  = acc0[v];
    orow[row + 16] = acc1[v];
    orow[row + 32] = acc2[v];
    orow[row + 48] = acc3[v];
  }
}

// ---------------------------------------------------------------------------
// Kernel 2: f1[i] = Wh[i,:]@a1 ; f2[i] = Wh[i,:]@a2
// ---------------------------------------------------------------------------
__global__ void __launch_bounds__(256)
f12_kernel(const float* __restrict__ Wh, const float* __restrict__ a,
           float* __restrict__ f1, float* __restrict__ f2) {
  const int i = blockIdx.x * blockDim.x + threadIdx.x;
  if (i >= N_NODES) return;
  const float* row = Wh + (size_t)i * OUT_F;
  float s1 = 0.f, s2 = 0.f;
#pragma unroll
  for (int n = 0; n < OUT_F; ++n) {
    const float w = row[n];
    s1 += w * a[n];
    s2 += w * a[OUT_F + n];
  }
  f1[i] = s1;
  f2[i] = s2;
}

// ---------------------------------------------------------------------------
// Kernel 3: gmax = max(f2)  (single block; shift bound for the softmax)
// ---------------------------------------------------------------------------
__global__ void __launch_bounds__(256)
fmax_kernel(const float* __restrict__ f2, float* __restrict__ gmax) {
  __shared__ float red[256];
  float m = -3.0e38f;
  for (int t = threadIdx.x; t < N_NODES; t += 256) m = fmaxf(m, f2[t]);
  red[threadIdx.x] = m;
  __syncthreads();
  for (int s = 128; s > 0; s >>= 1) {
    if ((int)threadIdx.x < s) red[threadIdx.x] = fmaxf(red[threadIdx.x], red[threadIdx.x + s]);
    __syncthreads();
  }
  if (threadIdx.x == 0) *gmax = red[0];
}

// ---------------------------------------------------------------------------
// Kernel 4: fused  mask + leaky_relu + softmax + (attention @ Wh) + elu.
// Block = 4 waves, owns 16 output rows. Wave w streams columns
// [w*2048, (w+1)*2048) in 16x64 adj tiles, double-buffered via
// global_load_async_to_lds_b128 (ASYNCcnt). P fragments are built in the
// WMMA A layout straight out of LDS; 4 WMMAs per K=4 chunk accumulate a
// 16x64 fp32 tile. Cross-wave combine through LDS at the end.
// ---------------------------------------------------------------------------
#define D_WAVES     4
#define JT          64
#define TILE_FLOATS (16 * JT)            // 1024 floats = 4096 B
#define COLS_PER_WAVE (N_NODES / D_WAVES)

__global__ void __launch_bounds__(D_WAVES * 32)
gat_attn_kernel(const float* __restrict__ adj, const float* __restrict__ Wh,
                const float* __restrict__ f1, const float* __restrict__ f2,
                const float* __restrict__ gmaxp, const int* __restrict__ do_att_p,
                float* __restrict__ out) {
  extern __shared__ float smem[];   // [0,8192): tiles (4 waves x 2 bufs x 1024)
                                    // reused as acc[4][16][64] after a barrier
                                    // [8192,8256): per-wave row sums [4][16]
  const int lane = threadIdx.x & 31;
  const int wave = threadIdx.x >> 5;
  const int r  = lane & 15;
  const int hh = lane >> 4;
  const int m0 = blockIdx.x * 16;
  const int do_att = *do_att_p;

  const float f1v  = f1[m0 + r];
  const float gmax = *gmaxp;
  const float smi  = f1v + gmax;
  const float mi   = fmaxf(smi, ALPHA * smi);   // upper bound on masked logits

  // async-copy addressing: instruction i copies rows (2i+hh), 16 bytes/lane
  const uint32_t ldsLane =
      (uint32_t)((wave * 2 * TILE_FLOATS) * sizeof(float)) + (uint32_t)lane * 16u;
  const uint64_t adjLaneBase =
      (uint64_t)(uintptr_t)adj +
      ((uint64_t)(m0 + hh) * N_NODES + (uint64_t)r * 4u) * sizeof(float);

  const int jbase  = wave * COLS_PER_WAVE;
  const int ntiles = COLS_PER_WAVE / JT;        // 32

  auto issue_tile = [&](int jt, int buf) {
    const uint64_t g = adjLaneBase + (uint64_t)jt * sizeof(float);
    const uint32_t l = ldsLane + (uint32_t)buf * (uint32_t)(TILE_FLOATS * sizeof(float));
#pragma unroll
    for (int i = 0; i < 8; ++i) {
      asm volatile("global_load_async_to_lds_b128 %0, %1, off"
                   :: "v"(l + (uint32_t)i * 512u),
                      "v"(g + (uint64_t)i * (2ull * N_NODES * sizeof(float)))
                   : "memory");
    }
  };

  v8f acc0 = {}, acc1 = {}, acc2 = {}, acc3 = {};
  float rsum = 0.f;

  issue_tile(jbase, 0);
  for (int t = 0; t < ntiles; ++t) {
    if (t + 1 < ntiles) {
      issue_tile(jbase + (t + 1) * JT, (t + 1) & 1);
      asm volatile("s_wait_asynccnt 8" ::: "memory");   // current tile landed
    } else {
      asm volatile("s_wait_asynccnt 0" ::: "memory");
    }
    const float* buf = smem + (size_t)(wave * 2 + (t & 1)) * TILE_FLOATS;
    const int jt = jbase + t * JT;

#pragma unroll 4
    for (int c = 0; c < 16; ++c) {
      const int j0 = jt + c * 4;
      const v2f adjv = *(const v2f*)(buf + r * JT + c * 4 + 2 * hh);
      const v2f f2v  = *(const v2f*)(f2 + j0 + 2 * hh);
      float p0, p1;
      if (do_att) {
        const float s0 = f1v + f2v.x, s1 = f1v + f2v.y;
        const float e0 = fmaxf(s0, ALPHA * s0);
        const float e1 = fmaxf(s1, ALPHA * s1);
        p0 = adjv.x > 0.f ? __expf(e0 - mi) : 0.f;
        p1 = adjv.y > 0.f ? __expf(e1 - mi) : 0.f;
      } else {
        p0 = adjv.x; p1 = adjv.y;
      }
      rsum += p0 + p1;
      v2f A; A.x = p0; A.y = p1;

      const float* wrow = Wh + (size_t)(j0 + 2 * hh) * OUT_F + r;
      v2f B0, B1, B2, B3;
      B0.x = wrow[0];  B0.y = wrow[OUT_F + 0];
      B1.x = wrow[16]; B1.y = wrow[OUT_F + 16];
      B2.x = wrow[32]; B2.y = wrow[OUT_F + 32];
      B3.x = wrow[48]; B3.y = wrow[OUT_F + 48];
      acc0 = wmma_f32(A, B0, acc0);
      acc1 = wmma_f32(A, B1, acc1);
      acc2 = wmma_f32(A, B2, acc2);
      acc3 = wmma_f32(A, B3, acc3);
    }
  }

  // per-wave row sums: lane l and l+16 together cover all columns of row m0+r
  const float rs = rsum + __shfl_xor(rsum, 16, 32);

  __syncthreads();   // all tile reads done; safe to reuse LDS for reduction

  // dump this wave's 16x64 accumulator tile (C layout -> row major) + row sums
  {
    float* accLds = smem + (size_t)wave * (16 * OUT_F);
#pragma unroll
    for (int v = 0; v < 8; ++v) {
      const int row = v + 8 * hh;
      accLds[row * OUT_F + r + 0]  = acc0[v];
      accLds[row * OUT_F + r + 16] = acc1[v];
      accLds[row * OUT_F + r + 32] = acc2[v];
      accLds[row * OUT_F + r + 48] = acc3[v];
    }
    if (lane < 16) smem[4 * 16 * OUT_F * D_WAVES / D_WAVES * 0 + 8192 + wave * 16 + r] = rs;
  }
  __syncthreads();

  // cross-wave combine, normalize, elu, store (coalesced)
  for (int idx = threadIdx.x; idx < 16 * OUT_F; idx += D_WAVES * 32) {
    const int row = idx >> 6;
    float v = smem[idx] + smem[16 * OUT_F + idx] + smem[2 * 16 * OUT_F + idx] +
              smem[3 * 16 * OUT_F + idx];
    float s = smem[8192 + row] + smem[8192 + 16 + row] + smem[8192 + 32 + row] +
              smem[8192 + 48 + row];
    const float inv = do_att ? (s > 0.f ? 1.f / s : 0.f) : 1.f;
    v *= inv;
    out[(size_t)(m0 + row) * OUT_F + (idx & 63)] = v > 0.f ? v : __expf(v) - 1.f;
  }
}

// ---------------------------------------------------------------------------
extern "C" void kernel_launch(void* const* d_in, const int* in_sizes, int n_in,
                              void* d_out, int out_size, void* d_ws, size_t ws_size,
                              hipStream_t stream) {
  const float* h_in  = (const float*)d_in[0];
  const float* adj   = (const float*)d_in[1];
  const float* W     = (const float*)d_in[2];
  const float* a_vec = (const float*)d_in[3];
  const int*   do_at = (const int*)d_in[4];
  float* out = (float*)d_out;

  // workspace: Wh (2 MB) | f1 (32 KB) | f2 (32 KB) | gmax (4 B)
  float* Wh   = (float*)d_ws;
  float* f1   = Wh + (size_t)N_NODES * OUT_F;
  float* f2   = f1 + N_NODES;
  float* gmax = f2 + N_NODES;

  wh_gemm_kernel<<<N_NODES / (16 * 8), 256, 0, stream>>>(h_in, W, Wh);
  f12_kernel<<<N_NODES / 256, 256, 0, stream>>>(Wh, a_vec, f1, f2);
  fmax_kernel<<<1, 256, 0, stream>>>(f2, gmax);

  const size_t shmem = 8192 * sizeof(float) + D_WAVES * 16 * sizeof(float); // 33 KB
  gat_attn_kernel<<<N_NODES / 16, D_WAVES * 32, shmem, stream>>>(
      adj, Wh, f1, f2, gmax, do_at, out);
}